// GraphAttentionLayer_3212635538174
// MI455X (gfx1250) — compile-verified
//
#include <hip/hip_runtime.h>
#include <hip/hip_bf16.h>

typedef __attribute__((ext_vector_type(2))) float v2f;
typedef __attribute__((ext_vector_type(8))) float v8f;

#define BN        64    // nodes per sample
#define NODE_DIM  128
#define EDGE_DIM  16
#define OUT_DIM   128
#define NHEADS    4
#define HEAD_DIM  32
#define LSTR      132   // padded LDS row stride (132 % 64 == 4 -> no bank conflicts)

__global__ __launch_bounds__(256, 2)
void gat_fused_kernel(const float* __restrict__ nodes,
                      const float* __restrict__ edgef,
                      const int*   __restrict__ mask,
                      const float* __restrict__ Wn,
                      const float* __restrict__ We,
                      const float* __restrict__ a_src,
                      const float* __restrict__ a_dst,
                      const float* __restrict__ Wo,
                      const float* __restrict__ bo,
                      float* __restrict__ out)
{
    __shared__ float Xs[BN * LSTR];          // node features (padded rows)
    __shared__ float Hs[BN * LSTR];          // projected features h
    __shared__ float sc[NHEADS * BN];        // attention scores / probabilities
    __shared__ float avec[HEAD_DIM];         // a_src + a_dst
    __shared__ float smax[NHEADS];
    __shared__ float sinv[NHEADS];
    __shared__ float wsum[OUT_DIM];

    const int t      = threadIdx.x;
    const int b      = blockIdx.x;
    const int lane   = t & 31;
    const int wave   = t >> 5;
    const int laneLo = lane & 15;
    const int laneHi = lane >> 4;

    // ---------------- Phase 1: stage node tile (64x128 fp32 = 32KB) into LDS
    {
        const float4* g4 = reinterpret_cast<const float4*>(nodes + (size_t)b * (BN * NODE_DIM));
#pragma unroll
        for (int j = 0; j < 8; ++j) {
            int    i = t + 256 * j;          // fully coalesced float4 index
            float4 v = g4[i];
            int    p   = i << 2;             // flat element index
            int    row = p >> 7;             // /128
            int    col = p & 127;
            float* d   = &Xs[row * LSTR + col];
            d[0] = v.x; d[1] = v.y; d[2] = v.z; d[3] = v.w;
        }
    }
    if (t < HEAD_DIM) avec[t] = a_src[t] + a_dst[t];
    __syncthreads();

    // ---------------- Phase 2: H = X @ W_node via V_WMMA_F32_16X16X4_F32
    // wave w -> column strip n0 = w*16, all 4 row tiles (B fragment reused x4)
    {
        const int n0 = wave * 16;
        v8f acc[4] = {};
        for (int k0 = 0; k0 < NODE_DIM; k0 += 4) {
            const int kb = k0 + laneHi * 2;  // lanes 0-15: K=k0,k0+1 ; lanes 16-31: K=k0+2,k0+3
            v2f bf;
            bf.x = Wn[kb * OUT_DIM + n0 + laneLo];
            bf.y = Wn[(kb + 1) * OUT_DIM + n0 + laneLo];
#pragma unroll
            for (int mt = 0; mt < 4; ++mt) {
                const v2f af = *reinterpret_cast<const v2f*>(&Xs[(mt * 16 + laneLo) * LSTR + kb]);
                acc[mt] = __builtin_amdgcn_wmma_f32_16x16x4_f32(
                    false, af, false, bf, (short)0, acc[mt], false, false);
            }
        }
        // C/D layout: VGPR v -> rows M=v (lanes 0-15) and M=v+8 (lanes 16-31), N = lane%16
#pragma unroll
        for (int mt = 0; mt < 4; ++mt) {
            const int rowb = mt * 16 + laneHi * 8;
#pragma unroll
            for (int v = 0; v < 8; ++v)
                Hs[(rowb + v) * LSTR + n0 + laneLo] = acc[mt][v];
        }
    }
    __syncthreads();

    // ---------------- Phase 3: attention scores, leaky relu, mask
    const int  hh    = t >> 6;   // head 0..3
    const int  nn    = t & 63;   // node 0..63
    float      s     = 0.0f;
    bool       valid;
    {
        const float* hr = &Hs[nn * LSTR + hh * HEAD_DIM];
#pragma unroll
        for (int d = 0; d < HEAD_DIM; ++d) s += hr[d] * avec[d];
        const float* er = edgef + ((size_t)b * BN + nn) * EDGE_DIM;
#pragma unroll
        for (int e = 0; e < EDGE_DIM; ++e) s += er[e] * We[e * NHEADS + hh];
        s     = (s > 0.0f) ? s : 0.2f * s;                // leaky_relu(0.2)
        valid = (mask[(size_t)b * BN + nn] != 0);
        sc[t] = valid ? s : -__builtin_inff();
    }
    __syncthreads();

    // ---------------- Phase 4: softmax reduction over N per head (tiny: 4 threads)
    if (t < NHEADS) {
        float mx = -__builtin_inff();
        for (int i = 0; i < BN; ++i) mx = fmaxf(mx, sc[t * BN + i]);
        float sum = 0.0f;
        for (int i = 0; i < BN; ++i) {
            float v = sc[t * BN + i];
            if (v > -__builtin_inff()) sum += __expf(v - mx);
        }
        smax[t] = mx;
        sinv[t] = (sum > 0.0f) ? (1.0f / sum) : 0.0f;
    }
    __syncthreads();
    sc[t] = valid ? __expf(s - smax[hh]) * sinv[hh] : 0.0f;
    __syncthreads();

    // ---------------- Phase 5: weighted = sum_n p[n,h] * H[n, o]
    if (t < OUT_DIM) {
        const int    ho = t >> 5;            // head of output channel
        const float* pw = &sc[ho * BN];
        float        acc = 0.0f;
        for (int n = 0; n < BN; ++n) acc = fmaf(pw[n], Hs[n * LSTR + t], acc);
        wsum[t] = acc;
    }
    __syncthreads();

    // ---------------- Phase 6: out = weighted @ W_out + b_out
    if (t < OUT_DIM) {
        float r = bo[t];
        for (int k = 0; k < OUT_DIM; ++k) r = fmaf(wsum[k], Wo[k * OUT_DIM + t], r);
        out[(size_t)b * OUT_DIM + t] = r;
    }
}

extern "C" void kernel_launch(void* const* d_in, const int* in_sizes, int n_in,
                              void* d_out, int out_size, void* d_ws, size_t ws_size,
                              hipStream_t stream) {
    const float* nodes = (const float*)d_in[0];
    const float* edgef = (const float*)d_in[1];
    const int*   mask  = (const int*)  d_in[2];
    const float* Wn    = (const float*)d_in[3];
    const float* We    = (const float*)d_in[4];
    const float* a_src = (const float*)d_in[5];
    const float* a_dst = (const float*)d_in[6];
    const float* Wo    = (const float*)d_in[7];
    const float* bo    = (const float*)d_in[8];
    float*       out   = (float*)d_out;

    const int B = in_sizes[0] / (BN * NODE_DIM);   // 4096
    gat_fused_kernel<<<B, 256, 0, stream>>>(nodes, edgef, mask, Wn, We, a_src, a_dst, Wo, bo, out);
}